// RoIHeads_62929860821258
// MI455X (gfx1250) — compile-verified
//
#include <hip/hip_runtime.h>
#include <stdint.h>
#include <math.h>

namespace {
constexpr int B_ = 8, P_ = 16384, CM1_ = 4;
constexpr int M_ = P_ * CM1_;        // 65536 candidates per image
constexpr int N_ = B_ * P_;          // 131072 proposals total
constexpr int K_ = 100;
constexpr float NEG_   = -1e30f;
constexpr float NEGH_  = -5e29f;     // NEG * 0.5 keep-threshold
constexpr float SCORE_TH_ = 0.05f;
constexpr float NMS_TH_   = 0.5f;
constexpr float XCLIP_ = 4.135166556742356f;  // log(1000/16)

// workspace layout (float offsets); every region 16B aligned
constexpr size_t WS_BOXES = 64;                              // [B*M*4]
constexpr size_t WS_MS    = WS_BOXES + (size_t)B_*M_*4;      // [B*M] masked scores
constexpr size_t WS_RAW   = WS_MS    + (size_t)B_*M_;        // [B*M] raw scores
constexpr size_t WS_SIDE  = WS_RAW   + (size_t)B_*M_;        // [B*M]
constexpr size_t WS_DXDY  = WS_SIDE  + (size_t)B_*M_;        // [B*M*3]
constexpr size_t WS_CONT  = WS_DXDY  + (size_t)B_*M_*3;      // [B*M]
constexpr size_t WS_OBOX  = WS_CONT  + (size_t)B_*M_;        // [B*M*4] offset boxes
} // namespace

__global__ void init_kernel(float* ws) {
  if (threadIdx.x < B_) ((unsigned*)ws)[threadIdx.x] = 0u;   // per-image coord max = 0.0f
}

__global__ void prep_kernel(const float* __restrict__ cls, const float* __restrict__ breg,
                            const float* __restrict__ hside, const float* __restrict__ dxm,
                            const float* __restrict__ cont, const float* __restrict__ props,
                            const int* __restrict__ hw, float* __restrict__ ws) {
  int n = blockIdx.x * blockDim.x + threadIdx.x;
  if (n >= N_) return;
  int b = n / P_, p = n % P_;

  float x1 = props[(size_t)n*4+0], y1 = props[(size_t)n*4+1];
  float x2 = props[(size_t)n*4+2], y2 = props[(size_t)n*4+3];
  float w = x2 - x1, h = y2 - y1;
  float cx = x1 + 0.5f*w, cy = y1 + 0.5f*h;

  float l[5], e[5];
  float mx = -INFINITY;
  #pragma unroll
  for (int c = 0; c < 5; ++c) { l[c] = cls[(size_t)n*5 + c]; mx = fmaxf(mx, l[c]); }
  float sum = 0.0f;
  #pragma unroll
  for (int c = 0; c < 5; ++c) { e[c] = expf(l[c] - mx); sum += e[c]; }

  float Wimg = (float)hw[1], Himg = (float)hw[0];
  float lmax = 0.0f;

  #pragma unroll
  for (int c = 1; c < 5; ++c) {
    const float* r = breg + (size_t)n*20 + (size_t)c*4;
    float dx = r[0] / 10.0f, dy = r[1] / 10.0f;
    float dw = fminf(r[2] / 5.0f, XCLIP_), dh = fminf(r[3] / 5.0f, XCLIP_);
    float pcx = dx*w + cx, pcy = dy*h + cy;
    float pw = expf(dw)*w, ph = expf(dh)*h;
    float bx1 = pcx - 0.5f*pw, by1 = pcy - 0.5f*ph;
    float bx2 = pcx + 0.5f*pw, by2 = pcy + 0.5f*ph;
    bx1 = fminf(fmaxf(bx1, 0.0f), Wimg);  bx2 = fminf(fmaxf(bx2, 0.0f), Wimg);
    by1 = fminf(fmaxf(by1, 0.0f), Himg);  by2 = fminf(fmaxf(by2, 0.0f), Himg);

    size_t g = (size_t)b*M_ + (size_t)p*CM1_ + (size_t)(c-1);
    ((float4*)(ws + WS_BOXES))[g] = make_float4(bx1, by1, bx2, by2);
    lmax = fmaxf(lmax, fmaxf(fmaxf(bx1, by1), fmaxf(bx2, by2)));

    float sc = e[c] / sum;
    ws[WS_RAW + g] = sc;
    ws[WS_MS  + g] = (sc > SCORE_TH_) ? sc : NEG_;
    ws[WS_SIDE + g] = (hside[(size_t)n*5 + c] > 0.0f) ? 1.0f : 0.0f;    // sigmoid(x)>0.5
    ws[WS_DXDY + g*3 + 0] = dxm[(size_t)n*15 + (size_t)c*3 + 0];
    ws[WS_DXDY + g*3 + 1] = dxm[(size_t)n*15 + (size_t)c*3 + 1];
    ws[WS_DXDY + g*3 + 2] = dxm[(size_t)n*15 + (size_t)c*3 + 2];
    const float* ct = cont + (size_t)n*25 + (size_t)c*5;
    float bvv = ct[0]; int bii = 0;
    #pragma unroll
    for (int j = 1; j < 5; ++j) if (ct[j] > bvv) { bvv = ct[j]; bii = j; }
    ws[WS_CONT + g] = (float)bii;
  }
  // per-image boxes.max(); coords >= 0 so uint-punned atomic max is monotone
  atomicMax((unsigned*)ws + b, __float_as_uint(lmax));
}

__global__ void offset_kernel(float* __restrict__ ws) {
  int g = blockIdx.x * blockDim.x + threadIdx.x;
  if (g >= B_ * M_) return;
  int b = g / M_, m = g % M_;
  float maxv = __uint_as_float(((unsigned*)ws)[b]);
  float off = (float)((m & 3) + 1) * (maxv + 1.0f);         // label * (max+1)
  float4 bx = ((const float4*)(ws + WS_BOXES))[g];
  ((float4*)(ws + WS_OBOX))[g] = make_float4(bx.x + off, bx.y + off, bx.z + off, bx.w + off);
}

__global__ __launch_bounds__(1024, 1)
void nms_kernel(const float* __restrict__ ws, float* __restrict__ out) {
  const int b = blockIdx.x;
  const int t = threadIdx.x;
  extern __shared__ float smem[];                 // 65536 masked scores (256 KB in LDS)
  float4* s4 = (float4*)smem;

  const float* f_ms   = ws + WS_MS   + (size_t)b * M_;
  const float* f_obox = ws + WS_OBOX + (size_t)b * M_ * 4;
  const float* f_box  = ws + WS_BOXES;
  const float* f_raw  = ws + WS_RAW;
  const float* f_side = ws + WS_SIDE;
  const float* f_dxdy = ws + WS_DXDY;
  const float* f_cont = ws + WS_CONT;

  __shared__ float rv[32];
  __shared__ int   ri[32];
  __shared__ float bb[5];      // prev selection: ox1 oy1 ox2 oy2 keep
  __shared__ int   sbi;        // prev best index (-1 = none yet)

  // ---- CDNA5 async bulk copy: 256KB masked scores global -> LDS, no VGPR round-trip ----
  unsigned lbase = (unsigned)(uintptr_t)smem;
  #pragma unroll
  for (int j = 0; j < 16; ++j) {
    int c = t + 1024 * j;                                    // float4 chunk index
    unsigned la = lbase + (unsigned)c * 16u;
    const float* gp = f_ms + (size_t)c * 4;
    asm volatile("global_load_async_to_lds_b128 %0, %1, off"
                 :: "v"(la), "v"((uint64_t)(uintptr_t)gp) : "memory");
  }
  if (t == 0) { bb[0]=0.f; bb[1]=0.f; bb[2]=0.f; bb[3]=0.f; bb[4]=0.f; sbi = -1; }
  asm volatile("s_wait_asynccnt 0" ::: "memory");
  __syncthreads();

  const int lane = t & 31, wid = t >> 5;

  float* o_boxes  = out;                         // [B,K,4]
  float* o_scores = out + B_*K_*4;               // [B,K]
  float* o_labels = o_scores + B_*K_;
  float* o_sides  = o_labels + B_*K_;
  float* o_dxdy   = o_sides  + B_*K_;            // [B,K,3]
  float* o_cont   = o_dxdy   + B_*K_*3;
  float* o_keep   = o_cont   + B_*K_;

  for (int k = 0; k < K_; ++k) {
    // ---- fused pass: apply suppression from selection k-1, then argmax for k ----
    const float ox1 = bb[0], oy1 = bb[1], ox2 = bb[2], oy2 = bb[3];
    const bool  pkeep = bb[4] != 0.0f;
    const int   pbest = sbi;
    const float a1 = (ox2 - ox1) * (oy2 - oy1);  // same expr as reference on obox[i]

    float bv = -INFINITY; int bi = 0;            // first-index tie-break = jnp.argmax
    #pragma unroll
    for (int j = 0; j < 16; ++j) {
      int c = t + 1024 * j;                      // thread-private chunk: no LDS hazards
      float4 v = s4[c];
      int base = c * 4;
      float sc[4] = { v.x, v.y, v.z, v.w };
      bool changed = false;
      #pragma unroll
      for (int q = 0; q < 4; ++q) {
        int m = base + q;
        float sv = sc[q];
        if (sv > NEGH_) {                        // skip dead entries -> shrinking L2 traffic
          if (m == pbest) { sv = NEG_; changed = true; }
          else if (pkeep) {
            float4 ob = ((const float4*)f_obox)[m];
            float ltx = fmaxf(ox1, ob.x), lty = fmaxf(oy1, ob.y);
            float rbx = fminf(ox2, ob.z), rby = fminf(oy2, ob.w);
            float iw = fmaxf(rbx - ltx, 0.0f), ih = fmaxf(rby - lty, 0.0f);
            float inter = iw * ih;
            float a2 = (ob.z - ob.x) * (ob.w - ob.y);
            float iou = inter / (a1 + a2 - inter);   // NaN (0/0) compares false, as in ref
            if (iou > NMS_TH_) { sv = NEG_; changed = true; }
          }
          sc[q] = sv;
        }
        if (sv > bv) { bv = sv; bi = m; }        // argmax on post-suppression value
      }
      if (changed) s4[c] = make_float4(sc[0], sc[1], sc[2], sc[3]);
    }
    #pragma unroll
    for (int off = 16; off > 0; off >>= 1) {     // wave32 shuffle tree
      float ov = __shfl_down(bv, off, 32);
      int   oi = __shfl_down(bi, off, 32);
      if (ov > bv || (ov == bv && oi < bi)) { bv = ov; bi = oi; }
    }
    if (lane == 0) { rv[wid] = bv; ri[wid] = bi; }
    __syncthreads();
    if (t < 32) {
      bv = rv[t]; bi = ri[t];
      #pragma unroll
      for (int off = 16; off > 0; off >>= 1) {
        float ov = __shfl_down(bv, off, 32);
        int   oi = __shfl_down(bi, off, 32);
        if (ov > bv || (ov == bv && oi < bi)) { bv = ov; bi = oi; }
      }
      if (t == 0) {
        bool keep = bv > NEGH_;                  // s[i] > NEG*0.5
        float kf = keep ? 1.0f : 0.0f;
        size_t g = (size_t)b * M_ + (size_t)bi;
        float4 ob = ((const float4*)f_obox)[bi];
        bb[0] = ob.x; bb[1] = ob.y; bb[2] = ob.z; bb[3] = ob.w;
        bb[4] = kf;
        sbi = bi;
        int slot = b * K_ + k;
        float4 box = ((const float4*)f_box)[g];
        o_boxes[slot*4+0] = box.x * kf;
        o_boxes[slot*4+1] = box.y * kf;
        o_boxes[slot*4+2] = box.z * kf;
        o_boxes[slot*4+3] = box.w * kf;
        o_scores[slot] = f_raw[g] * kf;
        o_labels[slot] = keep ? (float)((bi & 3) + 1) : 0.0f;
        o_sides[slot]  = f_side[g] * kf;
        o_dxdy[slot*3+0] = f_dxdy[g*3+0] * kf;
        o_dxdy[slot*3+1] = f_dxdy[g*3+1] * kf;
        o_dxdy[slot*3+2] = f_dxdy[g*3+2] * kf;
        o_cont[slot] = keep ? f_cont[g] : 0.0f;
        o_keep[slot] = kf;
      }
    }
    __syncthreads();

    // ---- early exit: scores are monotonically suppressed, so once nothing is
    // kept, every remaining iteration provably yields all-zero outputs ----
    if (bb[4] == 0.0f) {
      for (int kk = k + 1 + t; kk < K_; kk += 1024) {
        int slot = b * K_ + kk;
        o_boxes[slot*4+0] = 0.0f; o_boxes[slot*4+1] = 0.0f;
        o_boxes[slot*4+2] = 0.0f; o_boxes[slot*4+3] = 0.0f;
        o_scores[slot] = 0.0f;  o_labels[slot] = 0.0f;  o_sides[slot] = 0.0f;
        o_dxdy[slot*3+0] = 0.0f; o_dxdy[slot*3+1] = 0.0f; o_dxdy[slot*3+2] = 0.0f;
        o_cont[slot] = 0.0f;  o_keep[slot] = 0.0f;
      }
      break;                                    // uniform across workgroup (shared flag)
    }
  }
}

extern "C" void kernel_launch(void* const* d_in, const int* in_sizes, int n_in,
                              void* d_out, int out_size, void* d_ws, size_t ws_size,
                              hipStream_t stream) {
  (void)in_sizes; (void)n_in; (void)out_size; (void)ws_size;
  const float* cls  = (const float*)d_in[0];
  const float* breg = (const float*)d_in[1];
  const float* hs   = (const float*)d_in[2];
  const float* dxm  = (const float*)d_in[3];
  const float* ct   = (const float*)d_in[4];
  const float* pr   = (const float*)d_in[5];
  const int*   hw   = (const int*)d_in[6];
  float* ws  = (float*)d_ws;
  float* out = (float*)d_out;

  hipLaunchKernelGGL(init_kernel,   dim3(1),           dim3(64),  0, stream, ws);
  hipLaunchKernelGGL(prep_kernel,   dim3(N_ / 256),    dim3(256), 0, stream,
                     cls, breg, hs, dxm, ct, pr, hw, ws);
  hipLaunchKernelGGL(offset_kernel, dim3((B_*M_)/256), dim3(256), 0, stream, ws);
  hipLaunchKernelGGL(nms_kernel,    dim3(B_),          dim3(1024),
                     (size_t)M_ * sizeof(float), stream, ws, out);
}